// RXLayer_11355893531273
// MI455X (gfx1250) — compile-verified
//
#include <hip/hip_runtime.h>

typedef float v2f __attribute__((ext_vector_type(2)));
typedef float v8f __attribute__((ext_vector_type(8)));

#define DIM   4096
#define BATCH 64

// -------------------------------------------------------------------------
// Kernel 1: build the three 16x16 complex kron-factor matrices.
// Group g covers wires 4g..4g+3; within a group the first wire is the MSB.
// M[i,j] = (-i)^popcount(i^j) * prod_b (bit_b(i)==bit_b(j) ? cos : sin)
// theta ~ U[0,1), so the fast hardware sin/cos (v_sin_f32/v_cos_f32 with
// pre-scale by 1/2pi) are accurate — avoids the huge libm slow-path code.
// -------------------------------------------------------------------------
__global__ void build_gates(const float* __restrict__ theta,
                            float* __restrict__ gre,
                            float* __restrict__ gim) {
    int g = blockIdx.x;          // 0..2
    int t = threadIdx.x;         // 0..255
    int i = t >> 4;
    int j = t & 15;
    int x = i ^ j;
    float mag = 1.0f;
    #pragma unroll
    for (int b = 0; b < 4; ++b) {
        float th = theta[g * 4 + (3 - b)];   // bit b <-> wire (4g + 3 - b)
        mag *= ((x >> b) & 1) ? __sinf(th) : __cosf(th);
    }
    int p = __popc(x) & 3;                   // (-i)^p
    const float pre[4] = { 1.0f, 0.0f, -1.0f, 0.0f };
    const float pim[4] = { 0.0f, -1.0f, 0.0f, 1.0f };
    gre[g * 256 + t] = mag * pre[p];
    gim[g * 256 + t] = mag * pim[p];
}

// -------------------------------------------------------------------------
// Kernel 2: one tensor contraction pass.
//   out[i, n] = sum_k M[i,k] * in[k, n]   (complex), for each slice.
// Contracted axis has stride `rowStride` in the flat [4096*64] layout.
// slice = hi*16 + lo, base = hi*strideHi + lo*strideLo, 4 col-tiles of 16.
// One wave per (slice, coltile): 16x16x16 complex matmul = 16 f32 WMMAs.
// -------------------------------------------------------------------------
__global__ void __launch_bounds__(32)
rx_pass(const float* __restrict__ xr, const float* __restrict__ xi,
        float* __restrict__ outr,     float* __restrict__ outi,
        const float* __restrict__ gre, const float* __restrict__ gim,
        int rowStride, int strideHi, int strideLo) {
    const int lane   = threadIdx.x;
    const int laneLo = lane & 15;
    const int laneHi = lane >> 4;
    const int slice  = blockIdx.x >> 2;
    const int nb     = (blockIdx.x & 3) * 16;
    const int base   = (slice >> 4) * strideHi + (slice & 15) * strideLo;

    v8f accR = {};
    v8f accI = {};

    #pragma unroll
    for (int kb = 0; kb < 16; kb += 4) {
        const int k0 = kb + 2 * laneHi;          // this lane's K pair

        // A-fragment: 16x4 f32, lane%16 = M row, K = 2*(lane/16)+v
        v2f ar, ai, ain;
        ar.x = gre[laneLo * 16 + k0];
        ar.y = gre[laneLo * 16 + k0 + 1];
        ai.x = gim[laneLo * 16 + k0];
        ai.y = gim[laneLo * 16 + k0 + 1];
        ain  = -ai;                              // f32 WMMA has no A-neg modifier

        // B-fragment: 4x16 f32, mirrored (column-major load of X chunk)
        const float* pr = xr + base + k0 * rowStride + nb + laneLo;
        const float* pi = xi + base + k0 * rowStride + nb + laneLo;
        v2f br, bi;
        br.x = pr[0];  br.y = pr[rowStride];
        bi.x = pi[0];  bi.y = pi[rowStride];

        // complex MAC: Yr += Mr*Xr - Mi*Xi ; Yi += Mr*Xi + Mi*Xr
        accR = __builtin_amdgcn_wmma_f32_16x16x4_f32(false, ar,  false, br,
                                                     (short)0, accR, false, false);
        accR = __builtin_amdgcn_wmma_f32_16x16x4_f32(false, ain, false, bi,
                                                     (short)0, accR, false, false);
        accI = __builtin_amdgcn_wmma_f32_16x16x4_f32(false, ar,  false, bi,
                                                     (short)0, accI, false, false);
        accI = __builtin_amdgcn_wmma_f32_16x16x4_f32(false, ai,  false, br,
                                                     (short)0, accI, false, false);
    }

    // D layout: lane%16 = N col, VGPR r -> M row = r + 8*(lane/16)
    #pragma unroll
    for (int r = 0; r < 8; ++r) {
        const int row = r + 8 * laneHi;
        const int idx = base + row * rowStride + nb + laneLo;
        outr[idx] = accR[r];
        outi[idx] = accI[r];
    }
}

// -------------------------------------------------------------------------
// Host-side launch: gates -> pass C (low bits) -> pass B (mid) -> pass A (top)
// ws float layout: [0,768) gate re | [768,1536) gate im | pad |
//                  [2048,+262144) Yre | Yim | Zre | Zim   (~4.2 MB total)
// -------------------------------------------------------------------------
extern "C" void kernel_launch(void* const* d_in, const int* in_sizes, int n_in,
                              void* d_out, int out_size, void* d_ws, size_t ws_size,
                              hipStream_t stream) {
    const float* theta = (const float*)d_in[0];   // [12,1]
    const float* xr    = (const float*)d_in[1];   // [4096,64]
    const float* xi    = (const float*)d_in[2];   // [4096,64]
    float* out  = (float*)d_out;                  // [2,4096,64]
    float* W    = (float*)d_ws;

    float* GRE = W;
    float* GIM = W + 768;
    float* YR  = W + 2048;
    float* YI  = YR + DIM * BATCH;
    float* ZR  = YI + DIM * BATCH;
    float* ZI  = ZR + DIM * BATCH;
    float* OR_ = out;
    float* OI_ = out + DIM * BATCH;

    build_gates<<<3, 256, 0, stream>>>(theta, GRE, GIM);

    // Pass C: gate group 2 on low 4 bits. slice = j0*16+j1; rows stride 64.
    rx_pass<<<1024, 32, 0, stream>>>(xr, xi, YR, YI,
                                     GRE + 2 * 256, GIM + 2 * 256,
                                     64, 16384, 1024);
    // Pass B: gate group 1 on middle bits. slice = j0*16+i2; rows stride 1024.
    rx_pass<<<1024, 32, 0, stream>>>(YR, YI, ZR, ZI,
                                     GRE + 1 * 256, GIM + 1 * 256,
                                     1024, 16384, 64);
    // Pass A: gate group 0 on top bits. slice = i1*16+i2; rows stride 16384.
    rx_pass<<<1024, 32, 0, stream>>>(ZR, ZI, OR_, OI_,
                                     GRE, GIM,
                                     16384, 1024, 64);
}